// GalaxyTileDecoder_61203283968457
// MI455X (gfx1250) — compile-verified
//
#include <hip/hip_runtime.h>

// GalaxyTileDecoder fused kernel for gfx1250 (MI455X).
// Stage 1: decode GEMM (16 sources x 2601 px, K=8) via V_WMMA_F32_16X16X4_F32.
// Stage 2: constant-shift bilinear resample from LDS, float4 dual stores.

typedef float v2f __attribute__((ext_vector_type(2)));
typedef float v8f __attribute__((ext_vector_type(8)));

#define GAL_SLEN  51
#define GAL_PIX   (GAL_SLEN * GAL_SLEN)   // 2601
#define PT        52
#define PT_PIX    (PT * PT)               // 2704
#define PB        8                       // p-tiles per workgroup
#define NS        (PB * 2)                // sources per workgroup = 16 (== WMMA M)
#define NCHUNK    ((GAL_PIX + 15) / 16)   // 163 column chunks of 16
#define NG_TILE   (PT_PIX / 4)            // 676 float4 groups per tile
#define NG_ROW    (PT / 4)                // 13 float4 groups per row

__device__ __forceinline__ float tapv(const float* __restrict__ g, int r, int c) {
    // g points at a 51x51 decode image; (r,c) are coords in the 53x53 padded frame.
    // Valid (nonzero) region of the padded frame is [1,51]x[1,51]; everything else
    // (pad ring and grid_sample OOB) is zero.
    bool ok = (r >= 1) & (r <= GAL_SLEN) & (c >= 1) & (c <= GAL_SLEN);
    int rr = min(max(r - 1, 0), GAL_SLEN - 1);
    int cc = min(max(c - 1, 0), GAL_SLEN - 1);
    float v = g[rr * GAL_SLEN + cc];
    return ok ? v : 0.0f;
}

__global__ __launch_bounds__(256) void galaxy_tile_kernel(
    const float* __restrict__ locs,    // (P*2, 2)
    const float* __restrict__ z,       // (P*2, 8)
    const float* __restrict__ gbool,   // (P*2)
    const float* __restrict__ W,       // (8, 2601)
    const float* __restrict__ bvec,    // (2601)
    float* __restrict__ out,           // (2, P, 52, 52) flat
    int P)
{
    __shared__ float gal[NS * GAL_PIX];          // 16 x 2601 decode images (166 KB)
    __shared__ float s_fx[NS], s_fy[NS], s_bool[NS];
    __shared__ int   s_x0[NS], s_y0[NS];

    const int tid     = threadIdx.x;
    const int lane    = tid & 31;
    const int wave    = tid >> 5;
    const int s_base  = blockIdx.x * NS;
    const int total_s = P * 2;

    // ---- Stage 0: per-source shift constants ----------------------------------
    // Algebra: sample coord x_pix(j) = j + 2.5 - 4*locs[...,1] (unit-spacing grid),
    //          y_pix(i) = i + 2.5 - 4*locs[...,0], in the 53x53 padded frame.
    if (tid < NS) {
        int s = s_base + tid;
        float l0 = 0.0f, l1 = 0.0f, bb = 0.0f;
        if (s < total_s) {
            l0 = locs[s * 2 + 0];
            l1 = locs[s * 2 + 1];
            bb = gbool[s];
        }
        float xb = 2.5f - 4.0f * l1;
        float yb = 2.5f - 4.0f * l0;
        float x0 = floorf(xb), y0 = floorf(yb);
        s_x0[tid] = (int)x0;
        s_y0[tid] = (int)y0;
        s_fx[tid] = xb - x0;
        s_fy[tid] = yb - y0;
        s_bool[tid] = bb;
    }

    // A fragments for WMMA f32 16x16x4 (identical in every wave).
    // Layout: lane L holds row M=L%16; VGPR0 -> K = 2*(L/16), VGPR1 -> K+1.
    const int arow = lane & 15;
    const int kb   = (lane >> 4) * 2;
    const int sA   = s_base + arow;
    v2f a_lo = {0.0f, 0.0f}, a_hi = {0.0f, 0.0f};
    if (sA < total_s) {
        const float* zp = z + sA * 8;
        a_lo[0] = zp[kb + 0];
        a_lo[1] = zp[kb + 1];
        a_hi[0] = zp[kb + 4];
        a_hi[1] = zp[kb + 5];
    }

    __syncthreads();

    // ---- Stage 1: decode GEMM via WMMA f32 16x16x4, K=8 in two steps ----------
    for (int c = wave; c < NCHUNK; c += 8) {
        int col  = c * 16 + (lane & 15);
        int colc = col < GAL_PIX ? col : (GAL_PIX - 1);   // clamp, store guarded below
        // B layout mirrors A: lane L holds col N=L%16; VGPR0 -> K = 2*(L/16), VGPR1 -> K+1.
        v2f b_lo, b_hi;
        b_lo[0] = W[(kb + 0) * GAL_PIX + colc];
        b_lo[1] = W[(kb + 1) * GAL_PIX + colc];
        b_hi[0] = W[(kb + 4) * GAL_PIX + colc];
        b_hi[1] = W[(kb + 5) * GAL_PIX + colc];

        v8f acc = {0.0f, 0.0f, 0.0f, 0.0f, 0.0f, 0.0f, 0.0f, 0.0f};
        acc = __builtin_amdgcn_wmma_f32_16x16x4_f32(false, a_lo, false, b_lo,
                                                    (short)0, acc, false, false);
        acc = __builtin_amdgcn_wmma_f32_16x16x4_f32(false, a_hi, false, b_hi,
                                                    (short)0, acc, false, false);

        float bval  = bvec[colc];
        int   rbase = (lane >> 4) * 8;    // C/D layout: VGPR r -> M = r + 8*(L/16)
        if (col < GAL_PIX) {
            #pragma unroll
            for (int r = 0; r < 8; ++r) {
                int row = rbase + r;
                gal[row * GAL_PIX + col] = (acc[r] + bval) * s_bool[row];
            }
        }
    }

    __syncthreads();

    // ---- Stage 2: constant-shift bilinear resample, float4 output -------------
    // Each thread produces 4 consecutive pixels of one output row. The 4 pixels
    // share source rows (pr, pr+1) and need only 5 columns per row: 10 LDS taps
    // for 4 pixels instead of 16. Both tuple copies stored as b128.
    const long ooff = (long)P * PT_PIX;
    for (int g = tid; g < PB * NG_TILE; g += 256) {
        int t   = g / NG_TILE;
        int rem = g - t * NG_TILE;
        int i   = rem / NG_ROW;
        int j   = (rem - i * NG_ROW) * 4;
        int tglob = blockIdx.x * PB + t;

        float v0 = 0.0f, v1 = 0.0f, v2 = 0.0f, v3 = 0.0f;
        #pragma unroll
        for (int m = 0; m < 2; ++m) {
            int srow = t * 2 + m;
            int pr = s_y0[srow] + i;
            int pc = s_x0[srow] + j;
            float fx = s_fx[srow], fy = s_fy[srow];
            float w00 = (1.0f - fy) * (1.0f - fx);
            float w01 = (1.0f - fy) * fx;
            float w10 = fy * (1.0f - fx);
            float w11 = fy * fx;
            const float* gsrc = gal + srow * GAL_PIX;

            float top[5], bot[5];
            #pragma unroll
            for (int k = 0; k < 5; ++k) {
                top[k] = tapv(gsrc, pr,     pc + k);
                bot[k] = tapv(gsrc, pr + 1, pc + k);
            }
            v0 += w00 * top[0] + w01 * top[1] + w10 * bot[0] + w11 * bot[1];
            v1 += w00 * top[1] + w01 * top[2] + w10 * bot[1] + w11 * bot[2];
            v2 += w00 * top[2] + w01 * top[3] + w10 * bot[2] + w11 * bot[3];
            v3 += w00 * top[3] + w01 * top[4] + w10 * bot[3] + w11 * bot[4];
        }

        if (tglob < P) {
            float4 o4 = make_float4(v0, v1, v2, v3);
            long o = (long)tglob * PT_PIX + i * PT + j;   // 16B-aligned (j % 4 == 0)
            *(float4*)(out + o)        = o4;   // tuple element 0
            *(float4*)(out + o + ooff) = o4;   // tuple element 1 (identical)
        }
    }
}

extern "C" void kernel_launch(void* const* d_in, const int* in_sizes, int n_in,
                              void* d_out, int out_size, void* d_ws, size_t ws_size,
                              hipStream_t stream) {
    const float* locs    = (const float*)d_in[0];  // (P, 2, 2)
    const float* gparams = (const float*)d_in[1];  // (P, 2, 8)
    const float* gb      = (const float*)d_in[2];  // (P, 2, 1)
    const float* W       = (const float*)d_in[3];  // (8, 2601)
    const float* b       = (const float*)d_in[4];  // (2601)
    float* out = (float*)d_out;

    int P = in_sizes[0] / 4;                       // locs has P*2*2 elements
    int blocks = (P + PB - 1) / PB;
    galaxy_tile_kernel<<<blocks, 256, 0, stream>>>(locs, gparams, gb, W, b, out, P);
}